// MipNeRF_77867757076899
// MI455X (gfx1250) — compile-verified
//
#include <hip/hip_runtime.h>
#include <stdint.h>

// ---------------------------------------------------------------------------
// MipNeRF resample: 16384 rays, 129 bins. Memory-bound (~76MB -> ~3.3us @ 23.3TB/s).
// Design: 16 rays per 128-thread workgroup (4 waves, wave32).
//   Phase 0: TDM tensor_load_to_lds of t_vals (16x129) and weights (16x128) tiles.
//   Phase 1: weight blur -> w (all threads).
//   Phase 2: wave 0 computes 128-wide prefix sums for 16 rays via
//            V_WMMA_F32_16X16X4_F32 triangular matmuls (8 segments, carry chain).
//   Phase 3: CDF normalize. Phase 4: inverse-CDF sampling (LDS binary search).
//   Phase 5: cone->Gaussian means/covs, streamed out contiguously per thread.
// ---------------------------------------------------------------------------

typedef float v2f  __attribute__((ext_vector_type(2)));
typedef float v8f  __attribute__((ext_vector_type(8)));
typedef unsigned int u32x4 __attribute__((ext_vector_type(4)));
typedef int   i32x4 __attribute__((ext_vector_type(4)));
typedef int   i32x8 __attribute__((ext_vector_type(8)));

#define NBINS   129
#define NW      128
#define RPB     16     // rays per block
#define TPB     128    // threads per block

__global__ __launch_bounds__(TPB, 8)
void mipnerf_resample_kernel(const float* __restrict__ origins,
                             const float* __restrict__ directions,
                             const float* __restrict__ radii,
                             const float* __restrict__ t_vals,
                             const float* __restrict__ weights,
                             float* __restrict__ outT,
                             float* __restrict__ outM,
                             float* __restrict__ outC)
{
    __shared__ float t_lds[RPB * NBINS];     // staged t_vals tile (TDM target)
    __shared__ float wraw_lds[RPB * NW];     // staged weights tile (TDM target)
    __shared__ float w_lds[RPB * NBINS];     // blurred+padded weights (stride 129)
    __shared__ float cdf_lds[RPB * NBINS];   // S (scan) then CDF, stride 129
    __shared__ float nt_lds[RPB * NBINS];    // new t values
    __shared__ float rayO[RPB][3];
    __shared__ float rayD[RPB][3];
    __shared__ float rayR[RPB];
    __shared__ float invWsum[RPB];
    __shared__ float padStep[RPB];

    const int tid   = threadIdx.x;
    const int rbase = blockIdx.x * RPB;

    // ---------------- Phase 0: kick TDM DMA of the two tiles ----------------
#if __has_builtin(__builtin_amdgcn_tensor_load_to_lds)
    if (tid < 32) {
        // t_vals tile: 16 rows x 129 f32, row stride 129
        {
            unsigned long long ga = (unsigned long long)(const void*)(t_vals + (size_t)rbase * NBINS);
            unsigned lds = (unsigned)(unsigned long long)(uintptr_t)(void*)t_lds;
            u32x4 g0 = { 1u,                                   // count=1, user mode
                         lds,                                  // lds_addr
                         (unsigned)(ga & 0xFFFFFFFFu),         // global_addr[31:0]
                         (unsigned)((ga >> 32) & 0x1FFFFFFu) | 0x80000000u }; // addr[56:32] | type=2
            i32x8 g1 = { (int)0x00020000u,          // data_size=4B
                         (int)((unsigned)NBINS << 16), // tensor_dim0 low16 @ [63:48]
                         (int)((unsigned)RPB   << 16), // tensor_dim1 low16 @ [95:80]
                         (int)((unsigned)NBINS << 16), // tile_dim0 @ [127:112]
                         (int)RPB,                     // tile_dim1
                         (int)NBINS,                   // tensor_dim0_stride lo32
                         0, 0 };
            i32x4 z4 = {0,0,0,0};
            i32x8 z8 = {0,0,0,0,0,0,0,0};
            __builtin_amdgcn_tensor_load_to_lds(g0, g1, z4, z4, z8, 0);
        }
        // weights tile: 16 rows x 128 f32, row stride 128
        {
            unsigned long long ga = (unsigned long long)(const void*)(weights + (size_t)rbase * NW);
            unsigned lds = (unsigned)(unsigned long long)(uintptr_t)(void*)wraw_lds;
            u32x4 g0 = { 1u, lds,
                         (unsigned)(ga & 0xFFFFFFFFu),
                         (unsigned)((ga >> 32) & 0x1FFFFFFu) | 0x80000000u };
            i32x8 g1 = { (int)0x00020000u,
                         (int)((unsigned)NW  << 16),
                         (int)((unsigned)RPB << 16),
                         (int)((unsigned)NW  << 16),
                         (int)RPB,
                         (int)NW,
                         0, 0 };
            i32x4 z4 = {0,0,0,0};
            i32x8 z8 = {0,0,0,0,0,0,0,0};
            __builtin_amdgcn_tensor_load_to_lds(g0, g1, z4, z4, z8, 0);
        }
    }
#endif

    // Ray constants (overlap with DMA)
    if (tid < 48)       { int r = tid / 3, c = tid % 3; rayO[r][c] = origins[(rbase + r) * 3 + c]; }
    else if (tid < 96)  { int t2 = tid - 48; int r = t2 / 3, c = t2 % 3; rayD[r][c] = directions[(rbase + r) * 3 + c]; }
    else if (tid < 112) { int r = tid - 96; rayR[r] = radii[rbase + r]; }

#if __has_builtin(__builtin_amdgcn_tensor_load_to_lds)
    if (tid < 32) __builtin_amdgcn_s_wait_tensorcnt(0);
#else
    // Fallback: coalesced cooperative loads (tiles are contiguous in global).
    for (int i = tid; i < RPB * NBINS; i += TPB) t_lds[i]   = t_vals[(size_t)rbase * NBINS + i];
    for (int i = tid; i < RPB * NW;    i += TPB) wraw_lds[i] = weights[(size_t)rbase * NW + i];
#endif
    __syncthreads();

    // ---------------- Phase 1: blur weights -> w ----------------
    for (int idx = tid; idx < RPB * NW; idx += TPB) {
        int ray = idx >> 7, i = idx & 127;
        const float* wr = &wraw_lds[ray * NW];
        float wm1 = wr[i > 0 ? i - 1 : 0];
        float w0  = wr[i];
        float wp1 = wr[i < NW - 1 ? i + 1 : NW - 1];
        w_lds[ray * NBINS + i] = 0.5f * (fmaxf(wm1, w0) + fmaxf(w0, wp1)) + 0.01f;
    }
    __syncthreads();

    // ---------------- Phase 2: WMMA prefix scan (wave 0) ----------------
    // S[m][ray] = sum_{k<=m} w[k][ray], 128 elements as 8 segments of 16,
    // each segment = 4 chained V_WMMA_F32_16X16X4_F32 with lower-tri A.
    if (tid < 32) {
        const int lane = tid;
        const int m    = lane & 15;       // A row / D row-group, also B/D column
        const bool hiH = lane >= 16;
        float a0c[4], a1c[4];
#pragma unroll
        for (int c = 0; c < 4; ++c) {
            int k0 = 4 * c + (hiH ? 2 : 0);
            a0c[c] = (k0     <= m) ? 1.0f : 0.0f;
            a1c[c] = (k0 + 1 <= m) ? 1.0f : 0.0f;
        }
        const float* wrow = &w_lds[m * NBINS];  // column = lane&15
        float carry = 0.0f;
        const int mbase = hiH ? 8 : 0;
#pragma unroll 1
        for (int seg = 0; seg < 8; ++seg) {
            v8f acc;
#pragma unroll
            for (int j = 0; j < 8; ++j) acc[j] = carry;
#pragma unroll
            for (int c = 0; c < 4; ++c) {
                int kb = seg * 16 + 4 * c + (hiH ? 2 : 0);
                v2f a; a[0] = a0c[c]; a[1] = a1c[c];
                v2f b; b[0] = wrow[kb]; b[1] = wrow[kb + 1];
                acc = __builtin_amdgcn_wmma_f32_16x16x4_f32(
                        false, a, false, b, (short)0, acc, false, false);
            }
#pragma unroll
            for (int j = 0; j < 8; ++j)
                cdf_lds[m * NBINS + seg * 16 + mbase + j + 1] = acc[j];
            // carry for this column lives in D row 15: VGPR7 of lanes 16..31
            carry = __shfl(acc[7], (lane & 15) + 16, 32);
        }
    }
    __syncthreads();

    // ---------------- Phase 3: normalize to CDF ----------------
    if (tid < RPB) {
        float ws  = cdf_lds[tid * NBINS + NW];          // S[127] = sum(w)
        float pad = fmaxf(0.0f, 1e-5f - ws);
        invWsum[tid] = 1.0f / (ws + pad);
        padStep[tid] = pad * (1.0f / (float)NW);
    }
    __syncthreads();
    for (int idx = tid; idx < RPB * NBINS; idx += TPB) {
        int ray = idx / NBINS, k = idx - ray * NBINS;
        float v;
        if (k == 0)        v = 0.0f;
        else if (k == NW)  v = 1.0f;
        else               v = fminf(1.0f, (cdf_lds[idx] + (float)k * padStep[ray]) * invWsum[ray]);
        cdf_lds[idx] = v;
    }
    __syncthreads();

    // ---------------- Phase 4: inverse-CDF sampling ----------------
    const float USTEP = (1.0f - 1.1920928955078125e-07f) / 128.0f;
    for (int idx = tid; idx < RPB * NBINS; idx += TPB) {
        int ray = idx / NBINS, s = idx - ray * NBINS;
        float u = (float)s * USTEP;
        const float* crow = &cdf_lds[ray * NBINS];
        int lo = 0, hi = NBINS;                 // searchsorted right
        while (lo < hi) { int mid = (lo + hi) >> 1; if (crow[mid] <= u) lo = mid + 1; else hi = mid; }
        int i1 = lo < 1 ? 1 : (lo > NW ? NW : lo);
        int i0 = i1 - 1;
        float g0 = crow[i0], g1 = crow[i1];
        float b0 = t_lds[ray * NBINS + i0], b1 = t_lds[ray * NBINS + i1];
        float dn = g1 - g0;
        float tt = (u - g0) / dn;
        tt = (dn > 0.0f) ? tt : 0.0f;           // nan_to_num for 0/0
        tt = fminf(fmaxf(tt, 0.0f), 1.0f);
        float nt = b0 + tt * (b1 - b0);
        nt_lds[idx] = nt;
        outT[(size_t)(rbase + ray) * NBINS + s] = nt;
    }
    __syncthreads();

    // ---------------- Phase 5: cone -> Gaussian ----------------
    {
        const int ray  = tid >> 3;
        const int part = tid & 7;
        const int gray = rbase + ray;
        float d0 = rayD[ray][0], d1 = rayD[ray][1], d2 = rayD[ray][2];
        float o0 = rayO[ray][0], o1 = rayO[ray][1], o2 = rayO[ray][2];
        float r2 = rayR[ray] * rayR[ray];
        float q0 = d0 * d0, q1 = d1 * d1, q2 = d2 * d2;
        float invmag = 1.0f / (q0 + q1 + q2 + 1e-10f);
        float n0 = 1.0f - q0 * invmag;
        float n1 = 1.0f - q1 * invmag;
        float n2 = 1.0f - q2 * invmag;
        const float* nt = &nt_lds[ray * NBINS];
        for (int i = part * 16; i < part * 16 + 16; ++i) {
            float t0v = nt[i], t1v = nt[i + 1];
            float mu = 0.5f * (t0v + t1v), hw = 0.5f * (t1v - t0v);
            float mu2 = mu * mu, hw2 = hw * hw, hw4 = hw2 * hw2;
            float invden = 1.0f / (3.0f * mu2 + hw2);
            float t_mean = mu + 2.0f * mu * hw2 * invden;
            float t_var  = hw2 * (1.0f / 3.0f)
                         - (4.0f / 15.0f) * (hw4 * (12.0f * mu2 - hw2) * invden * invden);
            float r_var  = r2 * (0.25f * mu2 + (5.0f / 12.0f) * hw2
                         - (4.0f / 15.0f) * hw4 * invden);
            size_t base = ((size_t)gray * NW + i) * 3;
            outM[base + 0] = d0 * t_mean + o0;
            outM[base + 1] = d1 * t_mean + o1;
            outM[base + 2] = d2 * t_mean + o2;
            outC[base + 0] = t_var * q0 + r_var * n0;
            outC[base + 1] = t_var * q1 + r_var * n1;
            outC[base + 2] = t_var * q2 + r_var * n2;
        }
    }
}

extern "C" void kernel_launch(void* const* d_in, const int* in_sizes, int n_in,
                              void* d_out, int out_size, void* d_ws, size_t ws_size,
                              hipStream_t stream)
{
    (void)in_sizes; (void)n_in; (void)out_size; (void)d_ws; (void)ws_size;
    const float* origins    = (const float*)d_in[0];
    const float* directions = (const float*)d_in[1];
    const float* radii      = (const float*)d_in[2];
    const float* t_vals     = (const float*)d_in[3];
    const float* weights    = (const float*)d_in[4];

    const int B = 16384;
    float* outT = (float*)d_out;                       // (B,129)
    float* outM = outT + (size_t)B * NBINS;            // (B,128,3)
    float* outC = outM + (size_t)B * NW * 3;           // (B,128,3)

    dim3 grid(B / RPB);      // 1024 workgroups
    dim3 block(TPB);         // 128 threads = 4 waves (wave32)
    mipnerf_resample_kernel<<<grid, block, 0, stream>>>(
        origins, directions, radii, t_vals, weights, outT, outM, outC);
}